// Model12_9620726743225
// MI455X (gfx1250) — compile-verified
//
#include <hip/hip_runtime.h>
#include <math.h>

typedef float v2f __attribute__((ext_vector_type(2)));
typedef float v8f __attribute__((ext_vector_type(8)));

#define NNODE 20
#define GD    20
#define NEDGE 80
#define MNUM  50000
#define EPSV  1e-5f

// workspace float offsets
#define WS_CONTRIB 0      // 5 * 400
#define WS_ATK_WA 2000
#define WS_ATK_WE 2020
#define WS_ATK_B  2040
#define WS_TRF_WA 2060
#define WS_TRF_B  2080
#define WS_DEP_WA 2100
#define WS_DEP_B  2120
#define WS_D34    2140
#define WS_N4A    2160
#define WS_CW     2180
#define WS_C      2200
#define WS_TOTAL  2201

struct SmallArgs {
    const float* x1; const float* x2; const int* edges;
    const float* g1W[4]; const float* g1b[4]; const float* g1Wbeta;
    const float* g2W[4]; const float* g2b[4]; const float* g2Wbeta;
    const float* gate_W; const float* gate_b;
    const float* nn_W;   const float* nn_b;
    const float* lin1_W; const float* lin1_b;
    const float* atk_W;  const float* atk_b;
    const float* trf_W;  const float* trf_b;
    const float* dep_W;  const float* dep_b;
    const float* acc_W;  const float* acc_b;
    const float* n_w[4]; const float* n_b[4]; const float* n_a[4];
};

// column mapping: packed feature f (0..79) -> source column in weight matrix, -1 = zero pad
__device__ __forceinline__ int colmap(int mID, int f) {
    if (mID == 4) return (f < 65) ? f : -1;          // dep: x1|tmp|X directly
    int base;
    if (f < 15)      base = f;                        // x1 block
    else if (f < 25) base = 30 + (f - 15);            // tmp block (src)
    else if (f < 65) base = 50 + (f - 25);            // X block (src)
    else return -1;
    if (mID == 1 || mID == 3) {                       // dst variants shift into dst slots
        if (f < 15)      base += 15;
        else if (f < 25) base += 10;
        else             base += 40;
    }
    return base;
}

// TransformerConv + ReLU + graph_norm(axis=0). xin/xout live in LDS with given strides.
__device__ void tconv_gn(const float* xin, int istr, int IN,
                         const float* Wq, const float* bq,
                         const float* Wk, const float* bk,
                         const float* Wv, const float* bv,
                         const float* Ws, const float* bs,
                         const float* Wbeta,
                         const int* esrc, const int* edst,
                         const float* nw, const float* nb, const float* na,
                         float* xout, int ostr,
                         float* sc, int tid, int nt)
{
    float* q  = sc;
    float* kk = sc + 400;
    float* vv = sc + 800;
    float* sk = sc + 1200;
    float* att = sc + 1600;
    float* score = sc + 2000;

    // q,k,v,skip linears: 4 * 20 * 20 outputs
    for (int idx = tid; idx < 4 * NNODE * GD; idx += nt) {
        int mat = idx / 400; int rem = idx % 400;
        int i = rem / GD; int o = rem % GD;
        const float* W; const float* b; float* dst;
        if      (mat == 0) { W = Wq; b = bq; dst = q;  }
        else if (mat == 1) { W = Wk; b = bk; dst = kk; }
        else if (mat == 2) { W = Wv; b = bv; dst = vv; }
        else               { W = Ws; b = bs; dst = sk; }
        float acc = b[o];
        for (int f = 0; f < IN; ++f) acc += xin[i * istr + f] * W[o * IN + f];
        dst[rem] = acc;
    }
    __syncthreads();

    const float inv_sqrt_d = 0.22360679774997896f; // 1/sqrt(20)
    for (int e = tid; e < NEDGE; e += nt) {
        int s = esrc[e], d = edst[e];
        float acc = 0.f;
        for (int kf = 0; kf < GD; ++kf) acc += q[d * GD + kf] * kk[s * GD + kf];
        score[e] = acc * inv_sqrt_d;
    }
    __syncthreads();

    // per-destination softmax attention + gated skip + relu
    for (int n = tid; n < NNODE; n += nt) {
        float mx = -1e30f;
        for (int e = 0; e < NEDGE; ++e)
            if (edst[e] == n && score[e] > mx) mx = score[e];
        float z = 0.f;
        float o[GD];
        for (int kf = 0; kf < GD; ++kf) o[kf] = 0.f;
        for (int e = 0; e < NEDGE; ++e) {
            if (edst[e] != n) continue;
            float w = expf(score[e] - mx);
            z += w;
            int s = esrc[e];
            for (int kf = 0; kf < GD; ++kf) o[kf] += w * vv[s * GD + kf];
        }
        float invz = 1.f / z;
        float bsum = 0.f;
        for (int kf = 0; kf < GD; ++kf) {
            float ov = o[kf] * invz;
            float sv = sk[n * GD + kf];
            att[n * GD + kf] = ov;
            bsum += ov * Wbeta[kf] + sv * Wbeta[GD + kf] + (ov - sv) * Wbeta[2 * GD + kf];
        }
        float beta = 1.f / (1.f + expf(-bsum));
        for (int kf = 0; kf < GD; ++kf) {
            float h = beta * sk[n * GD + kf] + (1.f - beta) * att[n * GD + kf];
            xout[n * ostr + kf] = h > 0.f ? h : 0.f;
        }
    }
    __syncthreads();

    // graph_norm over nodes (axis 0)
    for (int kf = tid; kf < GD; kf += nt) {
        float mean = 0.f;
        for (int i = 0; i < NNODE; ++i) mean += xout[i * ostr + kf];
        mean *= (1.f / NNODE);
        float a = na[kf];
        float var = 0.f;
        for (int i = 0; i < NNODE; ++i) {
            float ov = xout[i * ostr + kf] - a * mean;
            var += ov * ov;
        }
        var *= (1.f / NNODE);
        float scl = nw[kf] * rsqrtf(var + EPSV);
        for (int i = 0; i < NNODE; ++i)
            xout[i * ostr + kf] = (xout[i * ostr + kf] - a * mean) * scl + nb[kf];
    }
    __syncthreads();
}

// ---------------- Kernel A: small graph + tmp + WMMA contribution tables ----------------
__global__ void __launch_bounds__(128)
kA(SmallArgs A, float* out0, float* ws)
{
    __shared__ float xcat[NNODE * 55];   // cols 0..14 x1 | 15..34 xa | 35..54 xb
    __shared__ float tmp10[NNODE * 10];
    __shared__ int   esrc[NEDGE], edst[NEDGE];
    __shared__ float sc[2200];
    __shared__ float F[32 * 80];
    __shared__ float Wp[80 * 32];

    const int tid = threadIdx.x;
    const int nt  = blockDim.x;

    for (int i = tid; i < NNODE * 15; i += nt) {
        int r = i / 15, c = i % 15;
        xcat[r * 55 + c] = A.x1[i];
    }
    for (int e = tid; e < NEDGE; e += nt) {
        esrc[e] = A.edges[e];
        edst[e] = A.edges[NEDGE + e];
    }
    __syncthreads();

    // layer 1: x1(15) -> xa (cols 15..34), norm n1
    tconv_gn(xcat, 55, 15,
             A.g1W[0], A.g1b[0], A.g1W[1], A.g1b[1], A.g1W[2], A.g1b[2], A.g1W[3], A.g1b[3],
             A.g1Wbeta, esrc, edst, A.n_w[0], A.n_b[0], A.n_a[0],
             xcat + 15, 55, sc, tid, nt);
    // layer 2: [x1|xa](35) -> xb (cols 35..54), norm n2
    tconv_gn(xcat, 55, 35,
             A.g2W[0], A.g2b[0], A.g2W[1], A.g2b[1], A.g2W[2], A.g2b[2], A.g2W[3], A.g2b[3],
             A.g2Wbeta, esrc, edst, A.n_w[1], A.n_b[1], A.n_a[1],
             xcat + 35, 55, sc, tid, nt);

    // gate softmax over nodes + nn pooling -> xg
    float* g  = sc;        // reuse scratch
    float* xg = sc + 32;
    for (int n = tid; n < NNODE; n += nt) {
        float acc = A.gate_b[0];
        for (int f = 0; f < 55; ++f) acc += xcat[n * 55 + f] * A.gate_W[f];
        g[n] = acc;
    }
    __syncthreads();
    for (int k = tid; k < GD; k += nt) {
        float mx = -1e30f;
        for (int n = 0; n < NNODE; ++n) mx = fmaxf(mx, g[n]);
        float zz = 0.f;
        for (int n = 0; n < NNODE; ++n) zz += expf(g[n] - mx);
        float acc = 0.f;
        for (int n = 0; n < NNODE; ++n) {
            float al = expf(g[n] - mx) / zz;
            float lin = A.nn_b[k];
            for (int f = 0; f < 55; ++f) lin += xcat[n * 55 + f] * A.nn_W[k * 55 + f];
            acc += al * lin;
        }
        xg[k] = acc;
    }
    __syncthreads();

    if (tid == 0) {
        // graph_norm n3 on 1-D xg, then V = tanh(lin1([relu(xg), x2]))
        float mean = 0.f;
        for (int k = 0; k < GD; ++k) mean += xg[k];
        mean *= (1.f / GD);
        float ov[GD]; float var = 0.f;
        for (int k = 0; k < GD; ++k) {
            ov[k] = xg[k] - A.n_a[2][k] * mean;
            var += ov[k] * ov[k];
        }
        var *= (1.f / GD);
        float inv = rsqrtf(var + EPSV);
        float acc = A.lin1_b[0];
        for (int k = 0; k < GD; ++k) {
            float xn = ov[k] * inv * A.n_w[2][k] + A.n_b[2][k];
            xn = xn > 0.f ? xn : 0.f;
            acc += xn * A.lin1_W[k];
        }
        for (int j = 0; j < 4; ++j) acc += A.x2[j] * A.lin1_W[20 + j];
        out0[0] = tanhf(acc);
    }

    // tmp: leave-one-out mean of col0 over mask groups
    for (int k = tid; k < 10; k += nt) {
        float s = 0.f, cnt = 0.f;
        for (int i = 0; i < NNODE; ++i) {
            float mk = (xcat[i * 55 + 5 + k] == 1.0f) ? 1.f : 0.f;
            s += mk * xcat[i * 55 + 0];
            cnt += mk;
        }
        float denom = (cnt > 1.f) ? (cnt - 1.f) : 1.f;
        for (int i = 0; i < NNODE; ++i) {
            float mk = (xcat[i * 55 + 5 + k] == 1.0f) ? 1.f : 0.f;
            tmp10[i * 10 + k] = (mk > 0.f) ? (s - xcat[i * 55 + 0]) / denom : 0.f;
        }
    }

    // scalar constants for kernel B
    for (int o = tid; o < GD; o += nt) {
        ws[WS_ATK_WA + o] = A.atk_W[o * 132 + 130];
        ws[WS_ATK_WE + o] = A.atk_W[o * 132 + 131];
        ws[WS_ATK_B  + o] = A.atk_b[o];
        ws[WS_TRF_WA + o] = A.trf_W[o * 131 + 130];
        ws[WS_TRF_B  + o] = A.trf_b[o];
        ws[WS_DEP_WA + o] = A.dep_W[o * 66 + 65];
        ws[WS_DEP_B  + o] = A.dep_b[o];
        ws[WS_D34    + o] = xcat[o * 55 + 3] + xcat[o * 55 + 4];
        ws[WS_N4A    + o] = A.n_a[3][o];
        ws[WS_CW     + o] = A.n_w[3][o] * A.acc_W[o];
    }
    if (tid == 0) {
        float C = A.acc_b[0];
        for (int k = 0; k < GD; ++k) C += A.n_b[3][k] * A.acc_W[k];
        ws[WS_C] = 24.f * C;
    }
    __syncthreads();

    // Build padded feature matrix F (32 x 80): x1 | tmp | X, zero pad
    for (int idx = tid; idx < 32 * 80; idx += nt) {
        int r = idx / 80, c = idx % 80;
        float v = 0.f;
        if (r < NNODE) {
            if (c < 15)      v = xcat[r * 55 + c];
            else if (c < 25) v = tmp10[r * 10 + (c - 15)];
            else if (c < 65) v = xcat[r * 55 + (c - 10)];   // X cols = xcat 15..54
        }
        F[idx] = v;
    }
    __syncthreads();

    // 5 contribution tables via V_WMMA_F32_16X16X4_F32 (wave 0)
    for (int mID = 0; mID < 5; ++mID) {
        const float* Wsrc = (mID < 2) ? A.atk_W : ((mID < 4) ? A.trf_W : A.dep_W);
        const int stride  = (mID < 2) ? 132 : ((mID < 4) ? 131 : 66);
        for (int idx = tid; idx < 80 * 32; idx += nt) {
            int f = idx / 32, o = idx % 32;
            float v = 0.f;
            if (o < GD) {
                int c = colmap(mID, f);
                if (c >= 0) v = Wsrc[o * stride + c];
            }
            Wp[idx] = v;   // Wp[f][o]
        }
        __syncthreads();

        if (tid < 32) {   // uniform per-wave predicate: EXEC all ones for wave 0
            const int lane = tid;
            const int hf   = lane >> 4;   // lane half selects K pair
            const int l16  = lane & 15;
            for (int mt = 0; mt < 2; ++mt) {
                for (int ntile = 0; ntile < 2; ++ntile) {
                    v8f acc = {0.f, 0.f, 0.f, 0.f, 0.f, 0.f, 0.f, 0.f};
                    for (int k0 = 0; k0 < 80; k0 += 4) {
                        v2f a, b;
                        int arow = mt * 16 + l16;
                        a.x = F[arow * 80 + k0 + 2 * hf + 0];
                        a.y = F[arow * 80 + k0 + 2 * hf + 1];
                        int bcol = ntile * 16 + l16;
                        b.x = Wp[(k0 + 2 * hf + 0) * 32 + bcol];
                        b.y = Wp[(k0 + 2 * hf + 1) * 32 + bcol];
                        acc = __builtin_amdgcn_wmma_f32_16x16x4_f32(
                                  false, a, false, b, (short)0, acc, false, false);
                    }
                    for (int r = 0; r < 8; ++r) {
                        int row = mt * 16 + r + hf * 8;
                        int col = ntile * 16 + l16;
                        if (row < NNODE && col < GD)
                            ws[WS_CONTRIB + mID * 400 + row * 20 + col] = acc[r];
                    }
                }
            }
        }
        __syncthreads();
    }
}

// ---------------- Kernel B: streaming order scoring -> p[m] ----------------
__global__ void __launch_bounds__(256)
kB(const int* __restrict__ a_src, const int* __restrict__ a_dst, const float* __restrict__ a_armies,
   const int* __restrict__ t_src, const int* __restrict__ t_dst, const float* __restrict__ t_armies,
   const int* __restrict__ d_tgt, const float* __restrict__ d_armies,
   const float* __restrict__ ws, float* __restrict__ outp)
{
    __shared__ float sw[WS_TOTAL];
    const int tid = threadIdx.x;
    for (int i = tid; i < WS_TOTAL; i += blockDim.x) sw[i] = ws[i];
    __syncthreads();

    const int m = blockIdx.x * blockDim.x + tid;
    if (m >= MNUM) return;

    float S1[20], S2[20];
#pragma unroll
    for (int k = 0; k < 20; ++k) { S1[k] = 0.f; S2[k] = 0.f; }

    int   ss[8], dd[8];
    float ar[8];

    // ---- attacks ----
    *(int4*)&ss[0] = ((const int4*)a_src)[2 * m];     *(int4*)&ss[4] = ((const int4*)a_src)[2 * m + 1];
    *(int4*)&dd[0] = ((const int4*)a_dst)[2 * m];     *(int4*)&dd[4] = ((const int4*)a_dst)[2 * m + 1];
    *(float4*)&ar[0] = ((const float4*)a_armies)[2 * m]; *(float4*)&ar[4] = ((const float4*)a_armies)[2 * m + 1];
    for (int a = 0; a < 8; ++a) {
        const float army  = ar[a];
        const float extra = 0.6f * army - 0.7f * sw[WS_D34 + dd[a]];
        const float* cs = sw + WS_CONTRIB + 0   + ss[a] * 20;
        const float* cd = sw + WS_CONTRIB + 400 + dd[a] * 20;
#pragma unroll
        for (int k = 0; k < 20; ++k) {
            float v = cs[k] + cd[k] + army * sw[WS_ATK_WA + k] + extra * sw[WS_ATK_WE + k] + sw[WS_ATK_B + k];
            S1[k] += v; S2[k] += v * v;
        }
    }

    // ---- transfers ----
    *(int4*)&ss[0] = ((const int4*)t_src)[2 * m];     *(int4*)&ss[4] = ((const int4*)t_src)[2 * m + 1];
    *(int4*)&dd[0] = ((const int4*)t_dst)[2 * m];     *(int4*)&dd[4] = ((const int4*)t_dst)[2 * m + 1];
    *(float4*)&ar[0] = ((const float4*)t_armies)[2 * m]; *(float4*)&ar[4] = ((const float4*)t_armies)[2 * m + 1];
    for (int a = 0; a < 8; ++a) {
        const float army = ar[a];
        const float* cs = sw + WS_CONTRIB + 800  + ss[a] * 20;
        const float* cd = sw + WS_CONTRIB + 1200 + dd[a] * 20;
#pragma unroll
        for (int k = 0; k < 20; ++k) {
            float v = cs[k] + cd[k] + army * sw[WS_TRF_WA + k] + sw[WS_TRF_B + k];
            S1[k] += v; S2[k] += v * v;
        }
    }

    // ---- deploys ----
    *(int4*)&ss[0] = ((const int4*)d_tgt)[2 * m];     *(int4*)&ss[4] = ((const int4*)d_tgt)[2 * m + 1];
    *(float4*)&ar[0] = ((const float4*)d_armies)[2 * m]; *(float4*)&ar[4] = ((const float4*)d_armies)[2 * m + 1];
    for (int a = 0; a < 8; ++a) {
        const float army = ar[a];
        const float* cg = sw + WS_CONTRIB + 1600 + ss[a] * 20;
#pragma unroll
        for (int k = 0; k < 20; ++k) {
            float v = cg[k] + army * sw[WS_DEP_WA + k] + sw[WS_DEP_B + k];
            S1[k] += v; S2[k] += v * v;
        }
    }

    // graph_norm(axis=1) + acc linear, closed form from S1/S2
    float acc = sw[WS_C];
#pragma unroll
    for (int k = 0; k < 20; ++k) {
        const float a4   = sw[WS_N4A + k];
        const float mean = S1[k] * (1.f / 24.f);
        const float sumsq = S2[k] - (2.f * a4 - a4 * a4) * mean * S1[k];
        const float var   = sumsq * (1.f / 24.f);
        acc += sw[WS_CW + k] * S1[k] * (1.f - a4) * rsqrtf(var + EPSV);
    }
    outp[1 + m] = acc;
}

// ---------------- Kernel C: in-place log_softmax over p ----------------
__global__ void __launch_bounds__(1024)
kC(float* __restrict__ out)
{
    __shared__ float red[1024];
    const int tid = threadIdx.x;

    float lm = -3.4e38f;
    for (int i = tid; i < MNUM; i += 1024) lm = fmaxf(lm, out[1 + i]);
    red[tid] = lm; __syncthreads();
    for (int s = 512; s > 0; s >>= 1) {
        if (tid < s) red[tid] = fmaxf(red[tid], red[tid + s]);
        __syncthreads();
    }
    const float mv = red[0];
    __syncthreads();

    float ls = 0.f;
    for (int i = tid; i < MNUM; i += 1024) ls += expf(out[1 + i] - mv);
    red[tid] = ls; __syncthreads();
    for (int s = 512; s > 0; s >>= 1) {
        if (tid < s) red[tid] += red[tid + s];
        __syncthreads();
    }
    const float shift = mv + logf(red[0]);

    for (int i = tid; i < MNUM; i += 1024) out[1 + i] -= shift;
}

extern "C" void kernel_launch(void* const* d_in, const int* in_sizes, int n_in,
                              void* d_out, int out_size, void* d_ws, size_t ws_size,
                              hipStream_t stream)
{
    (void)in_sizes; (void)n_in; (void)out_size; (void)ws_size;

    SmallArgs A;
    A.x1    = (const float*)d_in[0];
    A.x2    = (const float*)d_in[1];
    A.edges = (const int*)d_in[2];
    for (int j = 0; j < 4; ++j) {
        A.g1W[j] = (const float*)d_in[11 + 2 * j];
        A.g1b[j] = (const float*)d_in[12 + 2 * j];
        A.g2W[j] = (const float*)d_in[20 + 2 * j];
        A.g2b[j] = (const float*)d_in[21 + 2 * j];
    }
    A.g1Wbeta = (const float*)d_in[19];
    A.g2Wbeta = (const float*)d_in[28];
    A.gate_W = (const float*)d_in[29]; A.gate_b = (const float*)d_in[30];
    A.nn_W   = (const float*)d_in[31]; A.nn_b   = (const float*)d_in[32];
    A.lin1_W = (const float*)d_in[33]; A.lin1_b = (const float*)d_in[34];
    A.atk_W  = (const float*)d_in[35]; A.atk_b  = (const float*)d_in[36];
    A.trf_W  = (const float*)d_in[37]; A.trf_b  = (const float*)d_in[38];
    A.dep_W  = (const float*)d_in[39]; A.dep_b  = (const float*)d_in[40];
    A.acc_W  = (const float*)d_in[41]; A.acc_b  = (const float*)d_in[42];
    for (int j = 0; j < 4; ++j) {
        A.n_w[j] = (const float*)d_in[43 + 3 * j];
        A.n_b[j] = (const float*)d_in[44 + 3 * j];
        A.n_a[j] = (const float*)d_in[45 + 3 * j];
    }

    float* out = (float*)d_out;
    float* ws  = (float*)d_ws;

    kA<<<1, 128, 0, stream>>>(A, out, ws);

    const int threads = 256;
    const int blocks  = (MNUM + threads - 1) / threads;
    kB<<<blocks, threads, 0, stream>>>(
        (const int*)d_in[3], (const int*)d_in[4], (const float*)d_in[5],
        (const int*)d_in[6], (const int*)d_in[7], (const float*)d_in[8],
        (const int*)d_in[9], (const float*)d_in[10],
        ws, out);

    kC<<<1, 1024, 0, stream>>>(out);
}